// EmbeddingContextAttention_72507637891613
// MI455X (gfx1250) — compile-verified
//
#include <hip/hip_runtime.h>

// ---------------------------------------------------------------------------
// Reference reduces to:
//   att[b,k,l] = dot(sent[b, idx[b,k], :], sent[b,l,:])          (k = 0,1)
//   out[b,l, (k*3+c)*D + d] = att[b,k,l] * sent[b,l,d]           (c = 0,1,2)
// (CONTEXT_SIZE=3 -> offsets are [0,0,0]: three unshifted copies.)
//
// Traffic: ~48MB sent reads x2 + 302MB out writes -> store-bound (~17us @ 23.3TB/s).
// att GEMM (~50 MFLOP) done with V_WMMA_F32_16X16X4_F32 (fp32, matches reference).
// ---------------------------------------------------------------------------

typedef __attribute__((ext_vector_type(2))) float v2f;
typedef __attribute__((ext_vector_type(4))) float v4f;
typedef __attribute__((ext_vector_type(8))) float v8f;

#define BB 32
#define LL 512
#define DD 768
#define KC 64  // K-chunk staged in LDS per iteration

// ------------------------- Kernel 1: att via WMMA ---------------------------
// grid = B * (L/16) blocks, 32 threads (one wave) per block.
// Each wave computes one 16x16 att tile: rows = entities (only 0,1 valid),
// cols = 16 tokens; K=768 accumulated with 192 v_wmma_f32_16x16x4_f32.
__global__ __launch_bounds__(32)
void att_wmma_kernel(const float* __restrict__ sent,
                     const int* __restrict__ idxs,
                     float* __restrict__ att) {
  __shared__ __align__(16) float eLDS[2 * DD];    // both entity rows, full K
  __shared__ __align__(16) float sLDS[16 * KC];   // 16-token tile, K-chunk

  const int blk  = blockIdx.x;
  const int b    = blk >> 5;       // / (L/16)
  const int tile = blk & 31;       // % (L/16)
  const int lane = threadIdx.x;

  const float* sbase = sent + (size_t)b * LL * DD;
  const int i0 = idxs[b * 2 + 0];
  const int i1 = idxs[b * 2 + 1];

  // Stage both entity rows into LDS (coalesced v4f): 2*768/4 = 384 vec4.
  {
    const v4f* r0 = (const v4f*)(sbase + (size_t)i0 * DD);
    const v4f* r1 = (const v4f*)(sbase + (size_t)i1 * DD);
    v4f* e4 = (v4f*)eLDS;
    for (int i = lane; i < 2 * (DD / 4); i += 32) {
      int row = i / (DD / 4);
      int c4  = i - row * (DD / 4);
      e4[i] = (row == 0) ? r0[c4] : r1[c4];
    }
  }

  // Fragment geometry (32-bit WMMA 16x4 A / 4x16 B layout):
  //   lanes 0-15 : K = kk+0, kk+1  |  lanes 16-31 : K = kk+2, kk+3
  const int rowA   = lane & 15;     // A: M row   / B: N (token in tile)
  const int half   = lane >> 4;     // K half select
  const bool avalid = (rowA < 2);
  const int eoff   = (avalid ? rowA : 1) * DD + 2 * half;

  v8f c = {};
  for (int kb = 0; kb < DD; kb += KC) {
    __syncthreads();
    // Stage 16 rows x KC floats (coalesced): 256 vec4 -> 8 per lane.
    {
      v4f* s4 = (v4f*)sLDS;
      const int n4 = KC / 4;  // 16 vec4 per row
      for (int i = lane; i < 16 * n4; i += 32) {
        int r  = i / n4;
        int c4 = i - r * n4;
        s4[i] = ((const v4f*)(sbase + (size_t)(tile * 16 + r) * DD + kb))[c4];
      }
    }
    __syncthreads();

    for (int kk = 0; kk < KC; kk += 4) {
      v2f a = *(const v2f*)(eLDS + eoff + kb + kk);
      if (!avalid) { a.x = 0.0f; a.y = 0.0f; }   // value select, EXEC stays full
      v2f bf = *(const v2f*)(sLDS + rowA * KC + kk + 2 * half);
      c = __builtin_amdgcn_wmma_f32_16x16x4_f32(false, a, false, bf,
                                                (short)0, c, false, false);
    }
  }

  // D layout: VGPR0 = M=0 (lanes 0-15, N=lane); VGPR1 = M=1.
  if (lane < 16) {
    float* attb = att + (size_t)b * 2 * LL;
    attb[0 * LL + tile * 16 + lane] = c[0];
    attb[1 * LL + tile * 16 + lane] = c[1];
  }
}

// -------------------- Kernel 2: scale + replicate (streaming) ---------------
// grid = B*L blocks, 192 threads; each thread owns one vec4 of the D=768 row
// and emits 6 non-temporal vec4 stores (3 copies per entity).
__global__ __launch_bounds__(192)
void scale_concat_kernel(const float* __restrict__ sent,
                         const float* __restrict__ att,
                         float* __restrict__ out) {
  const int bl = blockIdx.x;            // b*L + l
  const int b  = bl / LL;
  const int l  = bl - b * LL;
  const int t  = threadIdx.x;           // 0..191

  const v4f  v  = ((const v4f*)(sent + (size_t)bl * DD))[t];
  const float a0 = att[(size_t)b * 2 * LL + l];
  const float a1 = att[(size_t)b * 2 * LL + LL + l];

  const v4f v0 = v * a0;
  const v4f v1 = v * a1;

  v4f* o = (v4f*)(out + (size_t)bl * 6 * DD) + t;
  const int s4 = DD / 4;  // 192 vec4 per D-segment
  __builtin_nontemporal_store(v0, o + 0 * s4);
  __builtin_nontemporal_store(v0, o + 1 * s4);
  __builtin_nontemporal_store(v0, o + 2 * s4);
  __builtin_nontemporal_store(v1, o + 3 * s4);
  __builtin_nontemporal_store(v1, o + 4 * s4);
  __builtin_nontemporal_store(v1, o + 5 * s4);
}

// ---------------------------------------------------------------------------
extern "C" void kernel_launch(void* const* d_in, const int* in_sizes, int n_in,
                              void* d_out, int out_size, void* d_ws, size_t ws_size,
                              hipStream_t stream) {
  const float* sent = (const float*)d_in[0];   // [B, L, D] f32
  const int*   idxs = (const int*)d_in[1];     // [B, 2]   i32
  float*       out  = (float*)d_out;           // [B, L, 6*D] f32
  float*       att  = (float*)d_ws;            // B*2*L floats = 128 KB scratch

  att_wmma_kernel<<<BB * (LL / 16), 32, 0, stream>>>(sent, idxs, att);
  scale_concat_kernel<<<BB * LL, 192, 0, stream>>>(sent, att, out);
}